// InvariantPointAttention_48756468744626
// MI455X (gfx1250) — compile-verified
//
#include <hip/hip_runtime.h>
#include <math.h>

// ---------------------------------------------------------------------------
// Invariant Point Attention for MI455X (gfx1250), f32 end-to-end.
// Memory-bound on pairwise_repr (151MB, read 2x; fits in 192MB L2).
// Matrix math via V_WMMA_F32_16X16X4_F32 (wave32); attn tile staged to LDS
// with GLOBAL_LOAD_ASYNC_TO_LDS_B128 (ASYNCcnt) in the res_pair kernel.
// ---------------------------------------------------------------------------

typedef __attribute__((ext_vector_type(2))) float v2f;
typedef __attribute__((ext_vector_type(8))) float v8f;

#define B_      2
#define N_      384
#define DIM_    384
#define H_      8
#define SK_     16
#define SV_     16
#define PK_     4
#define PV_     8
#define DP_     128
#define QKVW    768          // 128 qs | 128 ks | 128 vs | 96 qp | 96 kp | 192 vp
#define OUT_IN_ 1408         // 128 res_scalar | 192 res_pts | 64 res_norm | 1024 res_pair

#define SCALAR_SCALE_ 0.14433756729740643f   // (3*16)^-0.5
#define POINT_SCALE_  0.13608276348795434f   // (3*4*4.5)^-0.5
#define PAIR_SCALE_   0.57735026918962576f   // 3^-0.5
#define NEG_MAX_      (-3.402823466e38f)

__device__ __forceinline__ v8f wmma4(v2f a, v2f b, v8f c) {
  // D = A(16x4) * B(4x16) + C(16x16), f32
  return __builtin_amdgcn_wmma_f32_16x16x4_f32(false, a, false, b, (short)0, c,
                                               false, false);
}

// ---------------------------------------------------------------------------
// Generic f32 WMMA GEMM: C[M,N] = A[M,K] @ B[K,N] (+ bias[N]).
// One wave (32 threads) per 16x16 output tile. M,N,K multiples of 16/4.
// ---------------------------------------------------------------------------
__global__ __launch_bounds__(32)
void gemm_f32_wmma(const float* __restrict__ A, const float* __restrict__ Bm,
                   const float* __restrict__ bias, float* __restrict__ C,
                   int M, int K, int Nc)
{
  const int nt   = Nc >> 4;
  const int tile = blockIdx.x;
  const int ti = tile / nt, tj = tile % nt;
  const int lane = threadIdx.x, half = lane >> 4, lm = lane & 15;
  const int row = ti * 16 + lm;
  const int col = tj * 16 + lm;

  v8f c = {0.f,0.f,0.f,0.f,0.f,0.f,0.f,0.f};
  for (int k0 = 0; k0 < K; k0 += 4) {
    const int ka = k0 + 2 * half;
    v2f a, b;
    a.x = A[(size_t)row * K + ka];
    a.y = A[(size_t)row * K + ka + 1];
    b.x = Bm[(size_t)ka       * Nc + col];
    b.y = Bm[(size_t)(ka + 1) * Nc + col];
    c = wmma4(a, b, c);
  }
  const float bv = bias ? bias[col] : 0.f;
#pragma unroll
  for (int r = 0; r < 8; ++r) {
    const int orow = ti * 16 + r + 8 * half;
    C[(size_t)orow * Nc + col] = c[r] + bv;
  }
}

// ---------------------------------------------------------------------------
// Rigid apply: p_global = R @ p + t, in place on qkv point sections.
// One thread per (x,y,z) triple: 128 triples per (b,n) row.
// ---------------------------------------------------------------------------
__global__ __launch_bounds__(256)
void rigid_apply_k(float* __restrict__ qkv, const float* __restrict__ rot,
                   const float* __restrict__ trans)
{
  const int idx = blockIdx.x * blockDim.x + threadIdx.x;
  if (idx >= B_ * N_ * 128) return;
  const int bn = idx >> 7;          // (b,n)
  const int tq = idx & 127;         // triple within row
  float* p = qkv + (size_t)bn * QKVW + 384 + tq * 3;
  const float* R = rot + (size_t)bn * 9;
  const float* t = trans + (size_t)bn * 3;
  const float x = p[0], y = p[1], z = p[2];
  p[0] = R[0]*x + R[1]*y + R[2]*z + t[0];
  p[1] = R[3]*x + R[4]*y + R[5]*z + t[1];
  p[2] = R[6]*x + R[7]*y + R[8]*z + t[2];
}

// ---------------------------------------------------------------------------
// pair_bias[b,h,i,j] = (pairwise[b,i,j,:] . w_pair[:,h] + b_pair[h]) * PAIR_SCALE
// One wave per (i,j) row; coalesced float4 load (512B/wave/instr), shuffle reduce.
// First full stream of pairwise (151MB) -> warms L2 for res_pair pass.
// ---------------------------------------------------------------------------
__global__ __launch_bounds__(256)
void pair_bias_k(const float* __restrict__ pair, const float* __restrict__ w_pair,
                 const float* __restrict__ b_pair, float* __restrict__ pb)
{
  const int lane = threadIdx.x & 31;
  const int wid  = blockIdx.x * (blockDim.x >> 5) + (threadIdx.x >> 5);
  const int nw   = gridDim.x * (blockDim.x >> 5);

  float w[4][8];
#pragma unroll
  for (int kk = 0; kk < 4; ++kk)
#pragma unroll
    for (int h = 0; h < 8; ++h)
      w[kk][h] = w_pair[(4 * lane + kk) * 8 + h];
  float bp[8];
#pragma unroll
  for (int h = 0; h < 8; ++h) bp[h] = b_pair[h];

  const size_t total = (size_t)B_ * N_ * N_;
  for (size_t row = wid; row < total; row += nw) {
    const float4 p = ((const float4*)(pair + row * DP_))[lane];
    float acc[8];
#pragma unroll
    for (int h = 0; h < 8; ++h)
      acc[h] = p.x*w[0][h] + p.y*w[1][h] + p.z*w[2][h] + p.w*w[3][h];
#pragma unroll
    for (int off = 16; off > 0; off >>= 1)
#pragma unroll
      for (int h = 0; h < 8; ++h)
        acc[h] += __shfl_xor(acc[h], off, 32);
    if (lane < 8) {
      float v = 0.f, bsel = 0.f;
#pragma unroll
      for (int h = 0; h < 8; ++h)
        if (lane == h) { v = acc[h]; bsel = bp[h]; }
      const size_t bb  = row / ((size_t)N_ * N_);
      const size_t rem = row % ((size_t)N_ * N_);
      const size_t i = rem / N_, j = rem % N_;
      pb[((bb * H_ + lane) * N_ + i) * (size_t)N_ + j] = (v + bsel) * PAIR_SCALE_;
    }
  }
}

// ---------------------------------------------------------------------------
// Fused attention: one wave per (b, h, 16-row i-tile).
// logits = qs.ks*SS - c*(qn_i+kn_j) + 2c*qp.kp + pair_bias  (c = 0.5*pw*PS)
//   -> fused K=28 WMMA contraction with scales folded into q.
// Softmax over full row (LDS, stride 388 floats to avoid 64-bank conflicts),
// then WMMA attn@vs and attn@vp, rigid-inverse + point norms.
// ---------------------------------------------------------------------------
#define LSTR 388

__global__ __launch_bounds__(32)
void attn_k(const float* __restrict__ qkv, const float* __restrict__ pb,
            const float* __restrict__ rot, const float* __restrict__ trans,
            const float* __restrict__ pwts, const unsigned char* __restrict__ mask,
            float* __restrict__ attn_out, float* __restrict__ results)
{
  __shared__ float Lg[16 * LSTR];
  __shared__ float Qn[16];
  __shared__ float Inv[16];
  __shared__ float Pts[16 * 32];

  const int wg = blockIdx.x;
  const int it = wg % (N_ / 16);
  const int h  = (wg / (N_ / 16)) % H_;
  const int b  = wg / ((N_ / 16) * H_);
  const int i0 = it * 16;
  const int lane = threadIdx.x, half = lane >> 4, lm = lane & 15;

  const float pw   = log1pf(expf(pwts[h]));        // softplus
  const float cpt  = 0.5f * pw * POINT_SCALE_;
  const float twoc = 2.0f * cpt;

  // ||qp_i||^2 per row of the tile
  {
    const float* qp = qkv + (size_t)(b * N_ + i0 + lm) * QKVW + 384 + h * 12;
    float s = 0.f;
#pragma unroll
    for (int p = 0; p < 12; ++p) { const float v = qp[p]; s += v * v; }
    if (half == 0) Qn[lm] = s;
  }

  // A fragments: q-tilde (16 rows x 28 k): [qs*SS | qp_global*2c]
  v2f af[7];
  {
    const float* qb = qkv + (size_t)(b * N_ + i0 + lm) * QKVW;
#pragma unroll
    for (int g = 0; g < 7; ++g) {
      const int k0 = 4 * g + 2 * half, k1 = k0 + 1;
      af[g].x = (k0 < 16) ? qb[h*16 + k0] * SCALAR_SCALE_
                          : qb[384 + h*12 + (k0 - 16)] * twoc;
      af[g].y = (k1 < 16) ? qb[h*16 + k1] * SCALAR_SCALE_
                          : qb[384 + h*12 + (k1 - 16)] * twoc;
    }
  }
  __syncthreads();

  // ---- pass 1: logits into LDS ----
  for (int jt = 0; jt < N_ / 16; ++jt) {
    const int j0 = jt * 16;
    const float* kb = qkv + (size_t)(b * N_ + j0 + lm) * QKVW;
    v8f cc = {0.f,0.f,0.f,0.f,0.f,0.f,0.f,0.f};
#pragma unroll
    for (int g = 0; g < 7; ++g) {
      const int k0 = 4 * g + 2 * half, k1 = k0 + 1;
      v2f bf;
      bf.x = (k0 < 16) ? kb[128 + h*16 + k0] : kb[480 + h*12 + (k0 - 16)];
      bf.y = (k1 < 16) ? kb[128 + h*16 + k1] : kb[480 + h*12 + (k1 - 16)];
      cc = wmma4(af[g], bf, cc);
    }
    float kn = 0.f;
    {
      const float* kp = kb + 480 + h * 12;
#pragma unroll
      for (int p = 0; p < 12; ++p) { const float v = kp[p]; kn += v * v; }
    }
    const unsigned char mj = mask[b * N_ + j0 + lm];
#pragma unroll
    for (int r = 0; r < 8; ++r) {
      const int il = r + 8 * half;
      const int gi = i0 + il;
      float lg = cc[r] - cpt * (Qn[il] + kn)
               + pb[((size_t)(b * H_ + h) * N_ + gi) * N_ + j0 + lm];
      const unsigned char mi = mask[b * N_ + gi];
      lg = (mi && mj) ? lg : NEG_MAX_;
      Lg[il * LSTR + j0 + lm] = lg;
    }
  }
  __syncthreads();

  // ---- softmax (lanes 0..15 own one row each, float4 vectorized) ----
  if (lane < 16) {
    float4* row = (float4*)&Lg[lane * LSTR];
    float mx = NEG_MAX_;
    for (int t = 0; t < N_ / 4; ++t) {
      const float4 v = row[t];
      mx = fmaxf(mx, fmaxf(fmaxf(v.x, v.y), fmaxf(v.z, v.w)));
    }
    float s = 0.f;
    for (int t = 0; t < N_ / 4; ++t) {
      const float4 v = row[t];
      float4 e;
      e.x = expf(v.x - mx); e.y = expf(v.y - mx);
      e.z = expf(v.z - mx); e.w = expf(v.w - mx);
      s += e.x + e.y + e.z + e.w;
      row[t] = e;                       // keep unnormalized exp in LDS
    }
    Inv[lane] = 1.0f / s;
  }
  __syncthreads();

  // ---- write normalized attn to workspace (coalesced float4 stores) ----
  for (int il = 0; il < 16; ++il) {
    const float inv = Inv[il];
    const int gi = i0 + il;
    float4* dst = (float4*)(attn_out + ((size_t)(b * H_ + h) * N_ + gi) * N_);
    const float4* src = (const float4*)&Lg[il * LSTR];
    for (int t = lane; t < N_ / 4; t += 32) {
      const float4 e = src[t];
      float4 a; a.x = e.x*inv; a.y = e.y*inv; a.z = e.z*inv; a.w = e.w*inv;
      dst[t] = a;
    }
  }

  // ---- pass 2: res_scalar = attn@vs, res_pts(global) = attn@vp via WMMA ----
  v8f cs  = {0.f,0.f,0.f,0.f,0.f,0.f,0.f,0.f};
  v8f cp0 = {0.f,0.f,0.f,0.f,0.f,0.f,0.f,0.f};
  v8f cp1 = {0.f,0.f,0.f,0.f,0.f,0.f,0.f,0.f};
  const float invA = Inv[lm];
  for (int jt = 0; jt < N_ / 16; ++jt) {
    const int j0 = jt * 16;
#pragma unroll
    for (int ks = 0; ks < 4; ++ks) {
      const int kb0 = ks * 4 + 2 * half;
      v2f a;
      a.x = Lg[lm * LSTR + j0 + kb0]     * invA;
      a.y = Lg[lm * LSTR + j0 + kb0 + 1] * invA;
      const float* v0 = qkv + (size_t)(b * N_ + j0 + kb0) * QKVW;
      const float* v1 = v0 + QKVW;
      v2f bs;  bs.x  = v0[256 + h*16 + lm]; bs.y  = v1[256 + h*16 + lm];
      cs = wmma4(a, bs, cs);
      v2f bpv; bpv.x = v0[576 + h*24 + lm]; bpv.y = v1[576 + h*24 + lm];
      cp0 = wmma4(a, bpv, cp0);
      const int p = 16 + lm;
      v2f bq;
      bq.x = (p < 24) ? v0[576 + h*24 + p] : 0.f;
      bq.y = (p < 24) ? v1[576 + h*24 + p] : 0.f;
      cp1 = wmma4(a, bq, cp1);
    }
  }

  // ---- store res_scalar; dump points to LDS for cross-lane rigid inverse ----
#pragma unroll
  for (int r = 0; r < 8; ++r) {
    const int il = r + 8 * half;
    const int gi = i0 + il;
    results[(size_t)(b * N_ + gi) * OUT_IN_ + h * 16 + lm] = cs[r];
    Pts[il * 32 + lm]      = cp0[r];
    Pts[il * 32 + 16 + lm] = cp1[r];
  }
  __syncthreads();

  // ---- rigid inverse: local = R^T (g - t); write pts + norms ----
#pragma unroll
  for (int t4 = 0; t4 < 4; ++t4) {
    const int idx = t4 * 32 + lane;
    const int il = idx >> 3, d = idx & 7;
    const int gi = i0 + il;
    const float* R  = rot   + (size_t)(b * N_ + gi) * 9;
    const float* tr = trans + (size_t)(b * N_ + gi) * 3;
    const float gx = Pts[il * 32 + d * 3 + 0] - tr[0];
    const float gy = Pts[il * 32 + d * 3 + 1] - tr[1];
    const float gz = Pts[il * 32 + d * 3 + 2] - tr[2];
    const float lx = R[0]*gx + R[3]*gy + R[6]*gz;
    const float ly = R[1]*gx + R[4]*gy + R[7]*gz;
    const float lz = R[2]*gx + R[5]*gy + R[8]*gz;
    float* res = results + (size_t)(b * N_ + gi) * OUT_IN_;
    res[128 + h*24 + d*3 + 0] = lx;
    res[128 + h*24 + d*3 + 1] = ly;
    res[128 + h*24 + d*3 + 2] = lz;
    res[320 + h*8 + d] = sqrtf(lx*lx + ly*ly + lz*lz + 1e-8f);
  }
}

// ---------------------------------------------------------------------------
// res_pair[b,:,i,:] = attn[b,:,i,:] @ pairwise[b,i,:,:]   (M=8 heads pad 16,
// N=128, K=384). One block per (b,i), 8 waves x 16 columns.
// The 12KB attn tile is staged to LDS ONCE per block via async copy
// (GLOBAL_LOAD_ASYNC_TO_LDS_B128, ASYNCcnt) -- removes the 8x redundant,
// uncoalesced A-operand global traffic from the hot loop. Row stride 388
// floats (388%64==4) keeps the ds_load A-fragment reads bank-conflict-free.
// Second (L2-hot) stream of pairwise; B-fragment loads are 64B contiguous.
// ---------------------------------------------------------------------------
#define ASTRIDE 388

__global__ __launch_bounds__(256)
void res_pair_k(const float* __restrict__ attn, const float* __restrict__ pair,
                float* __restrict__ results)
{
  __shared__ float Attn[H_ * ASTRIDE];

  const int bi = blockIdx.x;
  const int b = bi / N_, i = bi % N_;
  const int wave = threadIdx.x >> 5, lane = threadIdx.x & 31;
  const int half = lane >> 4, lm = lane & 15;
  const int col0 = wave * 16;

  // stage attn[b, 0:8, i, 0:384] -> LDS (3 x 256 lanes x 16B, coalesced)
#pragma unroll
  for (int t = 0; t < 3; ++t) {
    const int idx  = t * 256 + threadIdx.x;     // 0..767 float4 chunks
    const int hrow = idx / 96;                  // head 0..7
    const int c4   = idx - hrow * 96;           // float4 column 0..95
    const float* gsrc = attn + ((size_t)(b * H_ + hrow) * N_ + i) * N_ + c4 * 4;
    const unsigned lds_dst =
        (unsigned)(uintptr_t)&Attn[hrow * ASTRIDE + c4 * 4];
    asm volatile("global_load_async_to_lds_b128 %0, %1, off"
                 :
                 : "v"(lds_dst), "v"(gsrc)
                 : "memory");
  }
  asm volatile("s_wait_asynccnt 0x0" ::: "memory");
  __syncthreads();

  const float* prow = pair + (size_t)(b * N_ + i) * N_ * DP_;  // [384 x 128]
  v8f c = {0.f,0.f,0.f,0.f,0.f,0.f,0.f,0.f};
  for (int k0 = 0; k0 < N_; k0 += 4) {
    const int kj = k0 + 2 * half;
    v2f a, bb;
    if (lm < 8) {
      a.x = Attn[lm * ASTRIDE + kj];
      a.y = Attn[lm * ASTRIDE + kj + 1];
    } else { a.x = 0.f; a.y = 0.f; }
    bb.x = prow[(size_t)kj       * DP_ + col0 + lm];
    bb.y = prow[(size_t)(kj + 1) * DP_ + col0 + lm];
    c = wmma4(a, bb, c);
  }
#pragma unroll
  for (int r = 0; r < 8; ++r) {
    const int m = r + 8 * half;          // head (rows 8..15 are padding)
    if (m < 8)
      results[(size_t)(b * N_ + i) * OUT_IN_ + 384 + m * DP_ + col0 + lm] = c[r];
  }
}

// ---------------------------------------------------------------------------
extern "C" void kernel_launch(void* const* d_in, const int* in_sizes, int n_in,
                              void* d_out, int out_size, void* d_ws, size_t ws_size,
                              hipStream_t stream)
{
  const float* x      = (const float*)d_in[0];   // [B,N,DIM]
  const float* rot    = (const float*)d_in[1];   // [B,N,3,3]
  const float* trans  = (const float*)d_in[2];   // [B,N,3]
  const float* pair   = (const float*)d_in[3];   // [B,N,N,DP]
  const unsigned char* mask = (const unsigned char*)d_in[4];  // [B,N] bool
  const float* w_qkv  = (const float*)d_in[5];   // [384,768]
  const float* pwts   = (const float*)d_in[6];   // [H]
  const float* w_pair = (const float*)d_in[7];   // [128,8]
  const float* b_pair = (const float*)d_in[8];   // [8]
  const float* w_out  = (const float*)d_in[9];   // [1408,384]
  const float* b_out  = (const float*)d_in[10];  // [384]
  float* out = (float*)d_out;                    // [B,N,DIM]

  char* ws = (char*)d_ws;
  float* ws_qkv  = (float*)(ws);                 // 768*768*4   = 2,359,296 B
  float* ws_pb   = (float*)(ws + 2359296);       // B*H*N*N*4   = 9,437,184 B
  float* ws_attn = (float*)(ws + 11796480);      // B*H*N*N*4   = 9,437,184 B
  float* ws_res  = (float*)(ws + 21233664);      // 768*1408*4  = 4,325,376 B

  // 1) qkv projection: [768,384]@[384,768]
  gemm_f32_wmma<<<48 * 48, 32, 0, stream>>>(x, w_qkv, nullptr, ws_qkv, 768, 384, QKVW);
  // 2) points -> global frame (in place)
  rigid_apply_k<<<(B_ * N_ * 128 + 255) / 256, 256, 0, stream>>>(ws_qkv, rot, trans);
  // 3) pair bias (streams pairwise once, warms L2)
  pair_bias_k<<<1024, 256, 0, stream>>>(pair, w_pair, b_pair, ws_pb);
  // 4) fused attention core
  attn_k<<<B_ * H_ * (N_ / 16), 32, 0, stream>>>(ws_qkv, ws_pb, rot, trans, pwts,
                                                 mask, ws_attn, ws_res);
  // 5) attn @ pairwise (streams pairwise again, ideally from L2)
  res_pair_k<<<B_ * N_, 256, 0, stream>>>(ws_attn, pair, ws_res);
  // 6) output projection: [768,1408]@[1408,384] + b_out
  gemm_f32_wmma<<<48 * 24, 32, 0, stream>>>(ws_res, w_out, b_out, out, 768, 1408, 384);
}